// RecurrentCritic_27109833572745
// MI455X (gfx1250) — compile-verified
//
#include <hip/hip_runtime.h>

// ---------------------------------------------------------------------------
// RecurrentCritic on MI455X (gfx1250, wave32, WMMA f32 16x16x4).
//
//   K0 prep_kernel  : interleave W_lstm[0:128,:] into k-pair-major layout so
//                     every WMMA B-fragment is one contiguous b64 load.
//   K1 front_kernel : yd = action @ W_act + b_act (WMMA); x=LN(yd), y=relu(yd);
//                     z_in = [x,y] @ W_lstm[0:128] + b_lstm (WMMA), stored in
//                     raw C-fragment layout keyed by (t, batch-tile, n-tile).
//   K2 lstm_kernel  : 16 WGs x 16 batch rows, T=512 steps. W_h B-fragments
//                     persistent in registers, h in LDS, c in registers;
//                     z = z_in(t) + h @ W_h via WMMA; gates; fused LN+W_out
//                     epilogue writes q directly (hs never materialized).
// ---------------------------------------------------------------------------

typedef __attribute__((ext_vector_type(2))) float v2f;
typedef __attribute__((ext_vector_type(8))) float v8f;

#define B_   256
#define T_   512
#define N_   (B_ * T_)
#define H_   64
#define EPSL 1e-12f

__device__ __forceinline__ v8f wmma4(v2f a, v2f b, v8f c) {
  // D = A(16x4 f32) * B(4x16 f32) + C(16x16 f32)
  return __builtin_amdgcn_wmma_f32_16x16x4_f32(
      /*neg_a=*/false, a, /*neg_b=*/false, b,
      /*c_mod=*/(short)0, c, /*reuse_a=*/false, /*reuse_b=*/false);
}

__device__ __forceinline__ float sigmoidf_(float x) {
  return 1.0f / (1.0f + __expf(-x));
}

// ===========================================================================
// Kernel 0: WxI[kp*512 + col*2 + j] = W_lstm[(2*kp+j)*256 + col]
// (k-pair interleave of the input half of W_lstm; 32768 floats)
// ===========================================================================
__global__ __launch_bounds__(256) void prep_kernel(
    const float* __restrict__ W_lstm, float* __restrict__ WxI)
{
  for (int d = blockIdx.x * 256 + threadIdx.x; d < 128 * 256;
       d += gridDim.x * 256) {
    const int kp  = d >> 9;
    const int rem = d & 511;
    const int col = rem >> 1;
    const int j   = rem & 1;
    WxI[d] = W_lstm[(2 * kp + j) * 256 + col];
  }
}

// ===========================================================================
// Kernel 1: one wave per 16-row tile; 4 waves / 128-thread block; 8192 tiles.
// ===========================================================================
__global__ __launch_bounds__(128) void front_kernel(
    const float* __restrict__ action,   // [N,64]
    const float* __restrict__ W_act,    // [64,64]
    const float* __restrict__ b_act,    // [64]
    const float* __restrict__ g2,       // [64]
    const float* __restrict__ beta2,    // [64]
    const float* __restrict__ WxI,      // interleaved W_lstm[0:128,:]
    const float* __restrict__ b_lstm,   // [256]
    float* __restrict__ zin)            // [T][16][16 ntiles][256] frag layout
{
  __shared__ float sW2[64 * 64];        // W_act, k-pair interleaved
  __shared__ float sXY[4][16 * 128];    // per-wave xy tile

  // interleave W_act while staging: sW2[kp*128 + col*2 + j] = W_act[(2kp+j)*64+col]
  for (int i = threadIdx.x; i < 64 * 64; i += 128) {
    const int kp  = i >> 7;
    const int rem = i & 127;
    const int col = rem >> 1;
    const int j   = rem & 1;
    sW2[i] = W_act[(2 * kp + j) * 64 + col];
  }
  __syncthreads();

  const int wv  = threadIdx.x >> 5;
  const int L   = threadIdx.x & 31;
  const int hi  = L >> 4;       // 0: K lanes {k0,k0+1}, 1: {k0+2,k0+3}
  const int l16 = L & 15;

  const int tile = blockIdx.x * 4 + wv; // 0..8191
  const int t    = tile >> 4;
  const int wgB  = tile & 15;

  // ---- yd[16,64] = action @ W_act ----
  v8f acc[4];
#pragma unroll
  for (int nt = 0; nt < 4; ++nt)
#pragma unroll
    for (int r = 0; r < 8; ++r) acc[nt][r] = 0.0f;

  const long rown = (long)(wgB * 16 + l16) * T_ + t;  // A-frag row (M = l16)
#pragma unroll
  for (int ks = 0; ks < 16; ++ks) {
    const int k  = 4 * ks + 2 * hi;
    const int kp = 2 * ks + hi;
    v2f a = *(const v2f*)(action + rown * 64 + k);
#pragma unroll
    for (int nt = 0; nt < 4; ++nt) {
      v2f b = *(const v2f*)&sW2[kp * 128 + (16 * nt + l16) * 2];
      acc[nt] = wmma4(a, b, acc[nt]);
    }
  }

  // ---- bias + per-row layernorm + relu -> sXY ----
  float ba[4], gg[4], bb[4];
#pragma unroll
  for (int nt = 0; nt < 4; ++nt) {
    ba[nt] = b_act[16 * nt + l16];
    gg[nt] = g2[16 * nt + l16];
    bb[nt] = beta2[16 * nt + l16];
  }
  float mu[8], rs[8];
#pragma unroll
  for (int r = 0; r < 8; ++r) {
    float s = 0.f, ss = 0.f;
#pragma unroll
    for (int nt = 0; nt < 4; ++nt) {
      float v = acc[nt][r] + ba[nt];
      acc[nt][r] = v;
      s += v; ss += v * v;
    }
#pragma unroll
    for (int m = 1; m < 16; m <<= 1) {
      s  += __shfl_xor(s,  m, 32);
      ss += __shfl_xor(ss, m, 32);
    }
    const float mm  = s * (1.0f / 64.0f);
    const float var = fmaxf(ss * (1.0f / 64.0f) - mm * mm, 0.0f);
    mu[r] = mm;
    rs[r] = rsqrtf(var + EPSL);
  }
#pragma unroll
  for (int nt = 0; nt < 4; ++nt) {
#pragma unroll
    for (int r = 0; r < 8; ++r) {
      const int m   = r + 8 * hi;
      const int col = 16 * nt + l16;
      const float v = acc[nt][r];
      sXY[wv][m * 128 + col]      = (v - mu[r]) * rs[r] * gg[nt] + bb[nt];  // x
      sXY[wv][m * 128 + 64 + col] = fmaxf(v, 0.0f);                        // y
    }
  }
  // same-wave LDS store->load is in-order; region is wave-private

  // ---- z[16,256] = xy @ W_lstm[0:128,:] + b_lstm ----
  v8f z[16];
#pragma unroll
  for (int nt = 0; nt < 16; ++nt) {
    const float bl = b_lstm[16 * nt + l16];
#pragma unroll
    for (int r = 0; r < 8; ++r) z[nt][r] = bl;
  }
  for (int ks = 0; ks < 32; ++ks) {
    const int k  = 4 * ks + 2 * hi;
    const int kp = 2 * ks + hi;
    v2f a = *(const v2f*)&sXY[wv][l16 * 128 + k];
    const float* wrow = WxI + (long)kp * 512 + l16 * 2;
#pragma unroll
    for (int nt = 0; nt < 16; ++nt) {
      v2f b = *(const v2f*)(wrow + 32 * nt);   // one contiguous b64 load
      z[nt] = wmma4(a, b, z[nt]);
    }
  }

  // ---- store fragments: contiguous 32B per lane ----
  float* zp = zin + ((long)(t * 16 + wgB) * 16) * 256;
#pragma unroll
  for (int nt = 0; nt < 16; ++nt) {
    *(v8f*)(zp + nt * 256 + L * 8) = z[nt];
  }
}

// ===========================================================================
// Kernel 2: recurrent LSTM + fused LN/W_out epilogue.
// 16 blocks x 256 threads; block wgB owns batch rows 16*wgB..+15.
// ===========================================================================
__global__ __launch_bounds__(256) void lstm_kernel(
    const float* __restrict__ c0,       // [256,64]
    const float* __restrict__ h0,       // [256,64]
    const float* __restrict__ W_lstm,   // [192,256] (rows 128..191 used)
    const float* __restrict__ g3,
    const float* __restrict__ beta3,
    const float* __restrict__ W_out,    // [64,1]
    const float* __restrict__ b_out,    // [1]
    const float* __restrict__ zin,      // frag layout from front_kernel
    float* __restrict__ out)            // q[N] | c_fin[256*64] | h_fin[256*64]
{
  __shared__ float sH[16 * 64];
  __shared__ float sZ[16 * 256];
  __shared__ float sWo[64], sG3[64], sB3[64];

  const int tid = threadIdx.x;
  const int wgB = blockIdx.x;
  const int wv  = tid >> 5;
  const int L   = tid & 31;
  const int hi  = L >> 4;
  const int l16 = L & 15;

  for (int i = tid; i < 16 * 64; i += 256)
    sH[i] = h0[(wgB * 16 + (i >> 6)) * 64 + (i & 63)];
  if (tid < 64) { sWo[tid] = W_out[tid]; sG3[tid] = g3[tid]; sB3[tid] = beta3[tid]; }
  const float bout = b_out[0];

  // gate-combine ownership: thread -> (rows rb..rb+3, col kcol)
  const int kcol = tid & 63;
  const int rb   = (tid >> 6) * 4;
  float c[4];
#pragma unroll
  for (int j = 0; j < 4; ++j)
    c[j] = c0[(wgB * 16 + rb + j) * 64 + kcol];

  // persistent B fragments of W_h = W_lstm[128:192,:]; wave owns 2 n-tiles
  const int nt0 = 2 * wv;
  v2f Bf0[16], Bf1[16];
#pragma unroll
  for (int ks = 0; ks < 16; ++ks) {
    const int k   = 128 + 4 * ks + 2 * hi;
    const int c0i = 16 * nt0 + l16;
    const int c1i = 16 * (nt0 + 1) + l16;
    Bf0[ks].x = W_lstm[k * 256 + c0i];
    Bf0[ks].y = W_lstm[(k + 1) * 256 + c0i];
    Bf1[ks].x = W_lstm[k * 256 + c1i];
    Bf1[ks].y = W_lstm[(k + 1) * 256 + c1i];
  }
  __syncthreads();

  for (int t = 0; t < T_; ++t) {
    const float* zp = zin + ((long)(t * 16 + wgB) * 16) * 256;
    v8f acc0 = *(const v8f*)(zp + nt0 * 256 + L * 8);         // z_in incl bias
    v8f acc1 = *(const v8f*)(zp + (nt0 + 1) * 256 + L * 8);
    if (t + 1 < T_) {  // warm caches for the next step's fragments
      const float* np = zin + ((long)((t + 1) * 16 + wgB) * 16) * 256;
      __builtin_prefetch(np + nt0 * 256 + L * 8, 0, 2);
      __builtin_prefetch(np + (nt0 + 1) * 256 + L * 8, 0, 2);
    }

    // z += h @ W_h  (A from LDS as b64, B persistent in registers)
#pragma unroll
    for (int ks = 0; ks < 16; ++ks) {
      const int k = 4 * ks + 2 * hi;
      v2f a = *(const v2f*)&sH[l16 * 64 + k];
      acc0 = wmma4(a, Bf0[ks], acc0);
      acc1 = wmma4(a, Bf1[ks], acc1);
    }

    // scatter z tile to LDS for gate combine
#pragma unroll
    for (int r = 0; r < 8; ++r) {
      const int row = r + 8 * hi;
      sZ[row * 256 + 16 * nt0 + l16]       = acc0[r];
      sZ[row * 256 + 16 * (nt0 + 1) + l16] = acc1[r];
    }
    __syncthreads();

    // gates (tf order i,j,f,o; forget bias 1.0), update c, write h to LDS
#pragma unroll
    for (int j = 0; j < 4; ++j) {
      const int row = rb + j;
      const float* zr = &sZ[row * 256];
      const float zi = zr[kcol];
      const float zj = zr[64 + kcol];
      const float zf = zr[128 + kcol];
      const float zo = zr[192 + kcol];
      const float cn = c[j] * sigmoidf_(zf + 1.0f) + sigmoidf_(zi) * tanhf(zj);
      c[j] = cn;
      sH[row * 64 + kcol] = sigmoidf_(zo) * tanhf(cn);
    }
    __syncthreads();

    // fused epilogue: q[b*T+t] = LN(h_row) . W_out + b_out
    {
      const int row = 2 * wv + hi;
      const float v0 = sH[row * 64 + l16];
      const float v1 = sH[row * 64 + l16 + 16];
      const float v2 = sH[row * 64 + l16 + 32];
      const float v3 = sH[row * 64 + l16 + 48];
      float s  = v0 + v1 + v2 + v3;
      float ss = v0 * v0 + v1 * v1 + v2 * v2 + v3 * v3;
#pragma unroll
      for (int m = 1; m < 16; m <<= 1) {
        s  += __shfl_xor(s,  m, 32);
        ss += __shfl_xor(ss, m, 32);
      }
      const float mm  = s * (1.0f / 64.0f);
      const float var = fmaxf(ss * (1.0f / 64.0f) - mm * mm, 0.0f);
      const float rsq = rsqrtf(var + EPSL);
      float q = ((v0 - mm) * rsq * sG3[l16]      + sB3[l16])      * sWo[l16]
              + ((v1 - mm) * rsq * sG3[l16 + 16] + sB3[l16 + 16]) * sWo[l16 + 16]
              + ((v2 - mm) * rsq * sG3[l16 + 32] + sB3[l16 + 32]) * sWo[l16 + 32]
              + ((v3 - mm) * rsq * sG3[l16 + 48] + sB3[l16 + 48]) * sWo[l16 + 48];
#pragma unroll
      for (int m = 1; m < 16; m <<= 1) q += __shfl_xor(q, m, 32);
      if (l16 == 0)
        out[(long)(wgB * 16 + row) * T_ + t] = q + bout;
    }
  }

  // finals: c_fin at N_, h_fin at N_ + 256*64
#pragma unroll
  for (int j = 0; j < 4; ++j)
    out[N_ + (wgB * 16 + rb + j) * 64 + kcol] = c[j];
  __syncthreads();
  for (int i = tid; i < 16 * 64; i += 256)
    out[N_ + B_ * H_ + (wgB * 16 + (i >> 6)) * 64 + (i & 63)] = sH[i];
}

// ===========================================================================
extern "C" void kernel_launch(void* const* d_in, const int* in_sizes, int n_in,
                              void* d_out, int out_size, void* d_ws, size_t ws_size,
                              hipStream_t stream) {
  (void)in_sizes; (void)n_in; (void)out_size; (void)ws_size;
  // setup_inputs order (obs branch is dead code in the reference -> unused):
  const float* action = (const float*)d_in[1];
  const float* c0     = (const float*)d_in[2];
  const float* h0     = (const float*)d_in[3];
  const float* W_act  = (const float*)d_in[8];
  const float* b_act  = (const float*)d_in[9];
  const float* g2     = (const float*)d_in[10];
  const float* beta2  = (const float*)d_in[11];
  const float* W_lstm = (const float*)d_in[12];
  const float* b_lstm = (const float*)d_in[13];
  const float* g3     = (const float*)d_in[14];
  const float* beta3  = (const float*)d_in[15];
  const float* W_out  = (const float*)d_in[16];
  const float* b_out  = (const float*)d_in[17];

  float* zin = (float*)d_ws;                 // N*256 f32 = 128 MiB
  float* WxI = zin + (size_t)N_ * 256;       // 128*256 f32 = 128 KiB
  float* out = (float*)d_out;                // q | c_fin | h_fin

  prep_kernel<<<32, 256, 0, stream>>>(W_lstm, WxI);
  front_kernel<<<2048, 128, 0, stream>>>(action, W_act, b_act, g2, beta2,
                                         WxI, b_lstm, zin);
  lstm_kernel<<<16, 256, 0, stream>>>(c0, h0, W_lstm, g3, beta3,
                                      W_out, b_out, zin, out);
}